// CenterSegment_56229711839544
// MI455X (gfx1250) — compile-verified
//
#include <hip/hip_runtime.h>
#include <stdint.h>

#define B_      8
#define C_      3
#define HG      256
#define WGD     256
#define HW      (HG * WGD)        // 65536
#define CHW     (C_ * HW)         // 196608
#define IMG     1024
#define K_      256
#define BOXSZ   64
#define HALF    32
#define PATCH   65
#define LPITCH  66                // LDS row pitch for ROI patch
#define THRESH  0.95f
#define NEGV    (-1e30f)
#define CAP     8192              // pow2 candidate cap (expected ~5.7K peaks/image)

// ---- output layout (floats, concatenated in reference return order) ----
#define PEAKS_OFF 0
#define BOXES_OFF (B_ * CHW)                                 // 1,572,864
#define POOL_OFF  (BOXES_OFF + B_ * K_ * 4)                  // 1,581,056
#define VALID_OFF (POOL_OFF + B_ * K_ * C_ * BOXSZ * BOXSZ)  // 26,746,880

__device__ __forceinline__ void async_g2l_b32(unsigned lds_addr, const float* gaddr) {
    // CDNA5 async copy: global -> LDS, per-lane, tracked by ASYNCcnt
    asm volatile("global_load_async_to_lds_b32 %0, %1, off"
                 :: "v"(lds_addr), "v"(gaddr) : "memory");
}
__device__ __forceinline__ void wait_async0() {
#if __has_builtin(__builtin_amdgcn_s_wait_asynccnt)
    __builtin_amdgcn_s_wait_asynccnt(0);
#else
    asm volatile("s_wait_asynccnt 0x0" ::: "memory");
#endif
}

// ---------------------------------------------------------------------------
__global__ void k_init(int* counts) {
    if (threadIdx.x < B_) counts[threadIdx.x] = 0;
}

// ---------------------------------------------------------------------------
// 5x5 'SAME' max-pool NMS via async-staged clamped 20x20 LDS tile
// (edge-clamp replication == -inf 'SAME' padding for max), separable pool.
__global__ void k_peaks(const float* __restrict__ grids,
                        float* __restrict__ out_peaks,
                        int* __restrict__ counts,
                        float* __restrict__ cand_val,
                        int* __restrict__ cand_idx) {
    __shared__ float tile[20 * 21];   // 20x20 halo tile, pitch 21 (bank-safe)
    __shared__ float hmax[20 * 16];   // row-direction 1x5 max

    int tid = threadIdx.x;
    int tx  = tid & 15;
    int ty  = tid >> 4;
    int blk   = blockIdx.x;           // (bc*16 + tileY)*16 + tileX
    int tilex = blk & 15;
    int tiley = (blk >> 4) & 15;
    int bc    = blk >> 8;             // b*C + c  (0..23)

    const float* plane = grids + (size_t)bc * HW;
    int x0 = tilex * 16 - 2;
    int y0 = tiley * 16 - 2;

    // async stage 20x20 clamped tile
    for (int e = tid; e < 400; e += 256) {
        int r  = e / 20;
        int cc = e - r * 20;
        int gy = y0 + r;  gy = gy < 0 ? 0 : (gy > HG - 1 ? HG - 1 : gy);
        int gx = x0 + cc; gx = gx < 0 ? 0 : (gx > WGD - 1 ? WGD - 1 : gx);
        async_g2l_b32((unsigned)(uintptr_t)&tile[r * 21 + cc],
                      plane + gy * WGD + gx);
    }
    wait_async0();
    __syncthreads();

    // separable pool: horizontal 1x5 max for all 20 rows x 16 output cols
    for (int e = tid; e < 320; e += 256) {
        int r  = e >> 4;
        int cc = e & 15;
        const float* p = &tile[r * 21 + cc];
        float h = fmaxf(fmaxf(fmaxf(p[0], p[1]), fmaxf(p[2], p[3])), p[4]);
        hmax[r * 16 + cc] = h;
    }
    __syncthreads();

    // vertical 5x1 max + NMS
    float v = tile[(ty + 2) * 21 + (tx + 2)];
    const float* h = &hmax[ty * 16 + tx];
    float m = fmaxf(fmaxf(fmaxf(h[0], h[16]), fmaxf(h[32], h[48])), h[64]);

    bool peak = (m == v) && (v > THRESH);
    int y = tiley * 16 + ty;
    int x = tilex * 16 + tx;
    out_peaks[(size_t)bc * HW + y * WGD + x] = peak ? v : 0.0f;
    if (peak) {
        int b = bc / C_;
        int r = (bc - b * C_) * HW + y * WGD + x;   // flat idx within image
        int p = atomicAdd(&counts[b], 1);
        if (p < CAP) {
            cand_val[b * CAP + p] = v;
            cand_idx[b * CAP + p] = r;
        }
    }
}

// ---------------------------------------------------------------------------
// One block per image: bitonic sort of packed keys (val desc, idx asc),
// zero-fill remainder with lowest non-peak flat indices, emit boxes/valid.
__global__ void k_topk(const int* __restrict__ counts,
                       const float* __restrict__ cand_val,
                       const int* __restrict__ cand_idx,
                       const float* __restrict__ peaks,
                       float* __restrict__ out_boxes,
                       float* __restrict__ out_valid,
                       int* __restrict__ sel_y1,
                       int* __restrict__ sel_x1,
                       int* __restrict__ sel_valid) {
    __shared__ unsigned long long keys[CAP];      // exactly 64 KB

    int b = blockIdx.x;
    int t = threadIdx.x;
    int nc = counts[b];
    if (nc > CAP) nc = CAP;

    const float* cv = cand_val + b * CAP;
    const int*   ci = cand_idx + b * CAP;

    // pack: key = (float_bits(val) << 32) | (0xFFFFFFFF - idx); pad with 0
    for (int i = t; i < CAP; i += 256) {
        if (i < nc) {
            keys[i] = ((unsigned long long)__float_as_uint(cv[i]) << 32) |
                      (unsigned)(0xFFFFFFFFu - (unsigned)ci[i]);
        } else {
            keys[i] = 0ull;
        }
    }
    __syncthreads();

    // bitonic sort, descending
    for (int k2 = 2; k2 <= CAP; k2 <<= 1) {
        for (int j = k2 >> 1; j > 0; j >>= 1) {
            for (int i = t; i < CAP; i += 256) {
                int ixj = i ^ j;
                if (ixj > i) {
                    unsigned long long a  = keys[i];
                    unsigned long long bb = keys[ixj];
                    bool desc = ((i & k2) == 0);
                    if (desc ? (a < bb) : (a > bb)) {
                        keys[i] = bb;
                        keys[ixj] = a;
                    }
                }
            }
            __syncthreads();
        }
    }

    // fill tail slots like top_k over the zero-padded array would:
    // value 0 at the lowest flat indices where peaks == 0
    int nsel = nc < K_ ? nc : K_;
    if (t == 0 && nsel < K_) {
        const float* pk = peaks + (size_t)b * CHW;
        int k = nsel;
        for (int flat = 0; flat < CHW && k < K_; ++flat) {
            if (pk[flat] == 0.0f) {
                keys[k] = (unsigned long long)(unsigned)(0xFFFFFFFFu - (unsigned)flat);
                ++k;
            }
        }
    }
    __syncthreads();

    {   // one slot per thread: decode + emit
        unsigned long long key = keys[t];
        unsigned vbits = (unsigned)(key >> 32);
        int idx = (int)(0xFFFFFFFFu - (unsigned)key);
        int rem = idx & (HW - 1);
        int cy  = rem >> 8;
        int cx  = rem & 255;
        float* bx = out_boxes + ((size_t)b * K_ + t) * 4;
        bx[0] = (float)(cx - HALF);
        bx[1] = (float)(cy - HALF);
        bx[2] = (float)(cx + HALF);
        bx[3] = (float)(cy + HALF);
        int vv = (vbits != 0u) ? 1 : 0;           // peak vals > 0.95 -> bits != 0
        out_valid[b * K_ + t] = (float)vv;
        sel_y1[b * K_ + t]    = cy - HALF;
        sel_x1[b * K_ + t]    = cx - HALF;
        sel_valid[b * K_ + t] = vv;
    }
}

// ---------------------------------------------------------------------------
// One block per (b,k): async-stage 3x65x65 patch (coords clamped at 0; only
// the negative side can be outside the 1024^2 image), then masked 2x2
// stride-1 max -> 3x64x64 outputs, NEG->0 rule, valid mask.
__global__ void k_roi(const float* __restrict__ images,
                      const int* __restrict__ sel_y1,
                      const int* __restrict__ sel_x1,
                      const int* __restrict__ sel_valid,
                      float* __restrict__ out_pool) {
    __shared__ float patch[C_ * PATCH * LPITCH];   // 3*65*66 floats = 51,480 B

    int blk = blockIdx.x;             // b*K + k
    int b   = blk >> 8;
    int t   = threadIdx.x;
    int y1  = sel_y1[blk];
    int x1  = sel_x1[blk];
    float validf = (float)sel_valid[blk];

    const float* imgb = images + (size_t)b * C_ * IMG * IMG;
    const int E = C_ * PATCH * PATCH;               // 12675 elements

    for (int e = t; e < E; e += 256) {
        int c   = e / (PATCH * PATCH);
        int rc  = e - c * (PATCH * PATCH);
        int r   = rc / PATCH;
        int col = rc - r * PATCH;
        int y = y1 + r;   if (y < 0) y = 0;         // clamp; masked at compute
        int x = x1 + col; if (x < 0) x = 0;
        async_g2l_b32((unsigned)(uintptr_t)&patch[(c * PATCH + r) * LPITCH + col],
                      imgb + ((size_t)c * IMG + y) * IMG + x);
    }
    wait_async0();
    __syncthreads();

    float* outp = out_pool + (size_t)blk * C_ * BOXSZ * BOXSZ;
    for (int o = t; o < C_ * BOXSZ * BOXSZ; o += 256) {
        int c   = o >> 12;
        int rc  = o & 4095;
        int r   = rc >> 6;
        int col = rc & 63;
        const float* p = &patch[(c * PATCH + r) * LPITCH + col];
        bool r0 = (y1 + r)       >= 0;
        bool r1 = (y1 + r + 1)   >= 0;
        bool c0 = (x1 + col)     >= 0;
        bool c1 = (x1 + col + 1) >= 0;
        float m = NEGV;
        if (r0 && c0) m = fmaxf(m, p[0]);
        if (r0 && c1) m = fmaxf(m, p[1]);
        if (r1 && c0) m = fmaxf(m, p[LPITCH]);
        if (r1 && c1) m = fmaxf(m, p[LPITCH + 1]);
        float out = (m < NEGV * 0.5f) ? 0.0f : m;   // all-outside bin -> 0
        outp[o] = out * validf;
    }
}

// ---------------------------------------------------------------------------
extern "C" void kernel_launch(void* const* d_in, const int* in_sizes, int n_in,
                              void* d_out, int out_size, void* d_ws, size_t ws_size,
                              hipStream_t stream) {
    const float* grids  = (const float*)d_in[0];   // [8,3,256,256]
    const float* images = (const float*)d_in[1];   // [8,3,1024,1024]

    float* out       = (float*)d_out;
    float* out_peaks = out + PEAKS_OFF;
    float* out_boxes = out + BOXES_OFF;
    float* out_pool  = out + POOL_OFF;
    float* out_valid = out + VALID_OFF;

    // workspace layout (32-bit words)
    int*   ws        = (int*)d_ws;
    int*   counts    = ws;                          // 8
    float* cand_val  = (float*)(ws + 8);            // 8*CAP
    int*   cand_idx  = ws + 8 + B_ * CAP;           // 8*CAP
    int*   sel_y1    = cand_idx + B_ * CAP;         // 8*256
    int*   sel_x1    = sel_y1 + B_ * K_;            // 8*256
    int*   sel_valid = sel_x1 + B_ * K_;            // 8*256

    k_init<<<1, 32, 0, stream>>>(counts);

    // 16x16 output tiles: 8*3*16*16 = 6144 blocks
    k_peaks<<<B_ * C_ * 16 * 16, 256, 0, stream>>>(grids, out_peaks, counts,
                                                   cand_val, cand_idx);

    k_topk<<<B_, 256, 0, stream>>>(counts, cand_val, cand_idx, out_peaks,
                                   out_boxes, out_valid,
                                   sel_y1, sel_x1, sel_valid);

    k_roi<<<B_ * K_, 256, 0, stream>>>(images, sel_y1, sel_x1, sel_valid,
                                       out_pool);
}